// PoolingModule_19808389169450
// MI455X (gfx1250) — compile-verified
//
#include <hip/hip_runtime.h>
#include <cstdint>
#include <cstddef>

// Problem constants (match reference)
#define BB   8
#define CC   256
#define HH   96
#define WW   96
#define HWX  (HH * WW)      // 9216 elements per channel plane
#define TAPS 9

// Tiling
#define TILE_W   32
#define TILE_H   8
#define NTHREADS 256        // 8 waves (wave32)
#define CG       64         // channels per block
#define NCG      (CC / CG)  // 4

// 8-byte pair with only 4-byte alignment guarantee (gather bases have arbitrary parity)
struct __attribute__((packed, aligned(4))) F2 { float x, y; };

__global__ __launch_bounds__(NTHREADS)
void deform_dw_conv_kernel(const float* __restrict__ x,
                           const float* __restrict__ off,
                           const float* __restrict__ wgt,
                           const float* __restrict__ bias,
                           float* __restrict__ out)
{
    // Offset staging area: [2*TAPS][NTHREADS] floats = 18 KB
    __shared__ float soff[2 * TAPS * NTHREADS];

    const int tid = threadIdx.x;
    const int tx0 = blockIdx.x * TILE_W;
    const int ty0 = blockIdx.y * TILE_H;
    const int bz  = blockIdx.z;
    const int b   = bz >> 2;            // NCG == 4
    const int c0  = (bz & 3) * CG;

    const int ox = tx0 + (tid & (TILE_W - 1));
    const int oy = ty0 + (tid >> 5);
    const int pix = oy * WW + ox;

    // ---------------------------------------------------------------
    // Phase 1: stage this tile's 18 offset planes into LDS with the
    // gfx1250 async global->LDS path (ASYNCcnt), no VGPRs consumed.
    // ---------------------------------------------------------------
    {
        const float* obase = off + (size_t)b * (2 * TAPS) * HWX;
        uint32_t voff = (uint32_t)(pix * 4);
        // Generic LDS pointer: low 32 bits are the wave-relative LDS byte address.
        uint32_t lptr = (uint32_t)(uintptr_t)(&soff[tid]);
#pragma unroll
        for (int k2 = 0; k2 < 2 * TAPS; ++k2) {
            asm volatile("global_load_async_to_lds_b32 %0, %1, %2"
                         :
                         : "v"(lptr), "v"(voff), "s"(obase)
                         : "memory");
            voff += HWX * 4;
            lptr += NTHREADS * 4;
        }
        asm volatile("s_wait_asynccnt 0" ::: "memory");
    }
    __syncthreads();

    // ---------------------------------------------------------------
    // Phase 2: per-pixel bilinear metadata, once, kept in REGISTERS and
    // reused across all CG channels. Each tap becomes two b64 pair-loads:
    //   pair r covers x[row_r, cx0] (.x) and x[row_r, cx0+1] (.y)
    // Out-of-bounds corners are handled by zeroed weights + clamped
    // indices; index collapses / buffer-end are fixed meta-side.
    // ---------------------------------------------------------------
    float wa[TAPS][2], wb[TAPS][2];   // weights applied to .x / .y of each pair
    int   ofs[TAPS][2];               // byte offsets of pair bases within a plane

#pragma unroll
    for (int t = 0; t < TAPS; ++t) {
        const int i = t / 3, j = t % 3;
        const float offy = soff[(2 * t) * NTHREADS + tid];
        const float offx = soff[(2 * t + 1) * NTHREADS + tid];
        const float sy = (float)(oy - 1 + i) + offy;   // PAD=1, DIL=1, STRIDE=1
        const float sx = (float)(ox - 1 + j) + offx;
        const float fy = floorf(sy), fx = floorf(sx);
        const int y0 = (int)fy, x0 = (int)fx;
        const float wy = sy - fy, wx = sx - fx;

        // validity folded into row/col factors
        const float fy0 = (y0 >= 0 && y0 < HH) ? (1.0f - wy) : 0.0f;
        const float fy1 = (y0 >= -1 && y0 < HH - 1) ? wy : 0.0f;
        const float fx0 = (x0 >= 0 && x0 < WW) ? (1.0f - wx) : 0.0f;
        const float fx1 = (x0 >= -1 && x0 < WW - 1) ? wx : 0.0f;

        const int cy0 = min(max(y0, 0), HH - 1);
        const int cy1 = min(max(y0 + 1, 0), HH - 1);
        const int cx0 = min(max(x0, 0), WW - 1);
        const int cx1 = min(max(x0 + 1, 0), WW - 1);

        float a0 = fy0 * fx0, b0 = fy0 * fx1;
        float a1 = fy1 * fx0, b1 = fy1 * fx1;

        // Column collapse (x0 == -1 or clamped): both corners map to x[cx0] = .x
        if (cx1 == cx0) { a0 += b0; b0 = 0.0f; a1 += b1; b1 = 0.0f; }

        int base0 = cy0 * WW + cx0;
        int base1 = cy1 * WW + cx0;
        // Buffer-end guard: a pair at the very last plane element would read
        // 4B past the tensor for (b=B-1,c=C-1). Shift base down one and move
        // the weight onto the .y lane (wb is provably 0 here after collapse).
        if (base0 == HWX - 1) { base0 -= 1; b0 = a0; a0 = 0.0f; }
        if (base1 == HWX - 1) { base1 -= 1; b1 = a1; a1 = 0.0f; }

        wa[t][0] = a0; wb[t][0] = b0;
        wa[t][1] = a1; wb[t][1] = b1;
        ofs[t][0] = base0 * 4;
        ofs[t][1] = base1 * 4;
    }

    // ---------------------------------------------------------------
    // Phase 3: channel loop. 18 b64 gathers + 45 FMAs per output.
    // Weights/bias addresses are block-uniform -> scalar loads.
    // ---------------------------------------------------------------
    const size_t bc0 = ((size_t)b * CC + c0) * (size_t)HWX;
    const char* xbase = (const char*)(x + bc0);

#pragma unroll 2
    for (int ci = 0; ci < CG; ++ci) {
        const char* xc = xbase + (size_t)ci * (HWX * 4);
        const float* wt = wgt + (size_t)(c0 + ci) * TAPS;

        // Prefetch next channel's plane near this thread's first sample (gfx1250 global_prefetch_b8)
        if (ci + 1 < CG)
            __builtin_prefetch(xc + (size_t)(HWX * 4) + ofs[0][0], 0, 1);

        float acc = bias[c0 + ci];
#pragma unroll
        for (int t = 0; t < TAPS; ++t) {
            F2 p0 = *(const F2*)(xc + ofs[t][0]);
            F2 p1 = *(const F2*)(xc + ofs[t][1]);
            float s = wa[t][0] * p0.x;
            s = fmaf(wb[t][0], p0.y, s);
            s = fmaf(wa[t][1], p1.x, s);
            s = fmaf(wb[t][1], p1.y, s);
            acc = fmaf(wt[t], s, acc);
        }
        out[bc0 + (size_t)ci * HWX + pix] = acc;
    }
}

extern "C" void kernel_launch(void* const* d_in, const int* in_sizes, int n_in,
                              void* d_out, int out_size, void* d_ws, size_t ws_size,
                              hipStream_t stream)
{
    (void)in_sizes; (void)n_in; (void)out_size; (void)d_ws; (void)ws_size;
    const float* x    = (const float*)d_in[0];
    const float* off  = (const float*)d_in[1];
    const float* wgt  = (const float*)d_in[2];
    const float* bias = (const float*)d_in[3];
    float* out = (float*)d_out;

    dim3 grid(WW / TILE_W, HH / TILE_H, BB * NCG);   // 3 x 12 x 32 = 1152 blocks
    dim3 block(NTHREADS);
    deform_dw_conv_kernel<<<grid, block, 0, stream>>>(x, off, wgt, bias, out);
}